// DGCNN_19353122635865
// MI455X (gfx1250) — compile-verified
//
#include <hip/hip_runtime.h>
#include <hip/hip_bf16.h>

typedef __attribute__((ext_vector_type(2))) float v2f;
typedef __attribute__((ext_vector_type(8))) float v8f;

#define N_NODES  100000
#define N_EDGES  1600000
#define N_GRAPHS 1000
#define NPG      100
#define HID      32
#define K_TOP    30
#define D_LAT    97

static inline int cdiv(long long a, int b) { return (int)((a + b - 1) / b); }

// ---------------- utility ----------------
__global__ void zero_f32(float* __restrict__ p, int n) {
    int i = blockIdx.x * blockDim.x + threadIdx.x;
    if (i < n) p[i] = 0.0f;
}

// ---------------- degree / norm ----------------
__global__ void deg_accum(const int* __restrict__ ei, const float* __restrict__ ew,
                          float* __restrict__ deg) {
    int e = blockIdx.x * blockDim.x + threadIdx.x;
    if (e < N_EDGES) atomicAdd(&deg[ei[N_EDGES + e]], ew[e]);
}

__global__ void deg_to_dinv(float* __restrict__ deg) {
    int i = blockIdx.x * blockDim.x + threadIdx.x;
    if (i < N_NODES) deg[i] = rsqrtf(deg[i] + 1.0f);   // deg+1 (self loop) > 0 always
}

// ---------------- embedding gather ----------------
__global__ void embed(const int* __restrict__ z, const float* __restrict__ zemb,
                      float* __restrict__ h) {
    int t = blockIdx.x * blockDim.x + threadIdx.x;
    if (t < N_NODES * HID) {
        int i = t >> 5, c = t & 31;
        h[t] = zemb[z[i] * HID + c];
    }
}

// ---------------- WMMA GEMM: xw[M,32] = h[M,32] @ W[32,32]  (f32 16x16x4) ----------------
// Requires M % 16 == 0 (true: 100000). A 16x4: lanes 0-15 -> M=lane, K=(0,1); lanes 16-31
// -> K=(2,3). B 4x16: lanes 0-15 -> N=lane, K=(0,1); lanes 16-31 -> K=(2,3).
// C 16x16: VGPR r -> row r (+8 for upper lane half), N = lane&15.
__global__ __launch_bounds__(256)
void gemm_n32_wmma(const float* __restrict__ h, const float* __restrict__ W,
                   float* __restrict__ xw, int M) {
    int wave = threadIdx.x >> 5;
    int lane = threadIdx.x & 31;
    int l15  = lane & 15;
    int half = lane >> 4;
    int rowBase = (blockIdx.x * 8 + wave) * 16;
    if (rowBase >= M) return;                 // wave-uniform: EXEC stays all-1s
    int rowA = rowBase + l15;                 // always < M since M % 16 == 0
    v8f c0 = {}; v8f c1 = {};
#pragma unroll
    for (int s = 0; s < 8; ++s) {
        int k = 4 * s + 2 * half;
        v2f a = *(const v2f*)(h + rowA * 32 + k);            // 8B-aligned (k even)
        v2f b0; b0.x = W[k * 32 + l15];       b0.y = W[(k + 1) * 32 + l15];
        v2f b1; b1.x = W[k * 32 + 16 + l15];  b1.y = W[(k + 1) * 32 + 16 + l15];
        c0 = __builtin_amdgcn_wmma_f32_16x16x4_f32(false, a, false, b0, (short)0, c0, false, false);
        c1 = __builtin_amdgcn_wmma_f32_16x16x4_f32(false, a, false, b1, (short)0, c1, false, false);
    }
#pragma unroll
    for (int r = 0; r < 8; ++r) {
        int row = rowBase + r + 8 * half;     // in range by construction: no guard
        xw[row * 32 + l15]      = c0[r];
        xw[row * 32 + 16 + l15] = c1[r];
    }
}

// ---------------- edge scatter (32 channels: one lane per channel) ----------------
// e is wave-uniform: force edge scalars through SGPRs so they lower to s_load
// (1 scalar fetch/wave instead of 32 broadcast lane loads per value).
__global__ void scatter32(const int* __restrict__ ei, const float* __restrict__ ew,
                          const float* __restrict__ dinv, const float* __restrict__ xw,
                          float* __restrict__ agg) {
    long long t = (long long)blockIdx.x * blockDim.x + threadIdx.x;
    int e = (int)(t >> 5);
    int c = (int)(t & 31);
    if (e < N_EDGES) {                         // uniform (e identical across the wave)
        int eu = __builtin_amdgcn_readfirstlane(e);
        int s  = ei[eu];
        int d  = ei[N_EDGES + eu];
        float norm = dinv[s] * ew[eu] * dinv[d];
        atomicAdd(&agg[(long long)d * 32 + c], norm * xw[(long long)s * 32 + c]);
    }
}

// ---------------- finalize: self-loop + bias + tanh; writes next h and x_cat slice ----
__global__ void finalize32(const float* __restrict__ agg, const float* __restrict__ xw,
                           const float* __restrict__ dinv, const float* __restrict__ b,
                           float* __restrict__ hout, float* __restrict__ xcat, int colbase) {
    int t = blockIdx.x * blockDim.x + threadIdx.x;
    if (t < N_NODES * 32) {
        int i = t >> 5, c = t & 31;
        float di = dinv[i];
        float v = tanhf(agg[t] + xw[t] * di * di + b[c]);
        hout[t] = v;
        xcat[(long long)i * D_LAT + colbase + c] = v;
    }
}

// ---------------- layer 4 (32 -> 1) ----------------
__global__ void gemv_out1(const float* __restrict__ h, const float* __restrict__ W4,
                          float* __restrict__ xw1) {
    int i = blockIdx.x * blockDim.x + threadIdx.x;
    if (i < N_NODES) {
        float s = 0.0f;
#pragma unroll
        for (int k = 0; k < 32; ++k) s += h[i * 32 + k] * W4[k];
        xw1[i] = s;
    }
}

__global__ void scatter1(const int* __restrict__ ei, const float* __restrict__ ew,
                         const float* __restrict__ dinv, const float* __restrict__ xw1,
                         float* __restrict__ agg1) {
    int e = blockIdx.x * blockDim.x + threadIdx.x;
    if (e < N_EDGES) {
        int s = ei[e], d = ei[N_EDGES + e];
        atomicAdd(&agg1[d], dinv[s] * ew[e] * dinv[d] * xw1[s]);
    }
}

__global__ void finalize1(const float* __restrict__ agg1, const float* __restrict__ xw1,
                          const float* __restrict__ dinv, const float* __restrict__ b4,
                          float* __restrict__ xcat) {
    int i = blockIdx.x * blockDim.x + threadIdx.x;
    if (i < N_NODES) {
        float di = dinv[i];
        xcat[(long long)i * D_LAT + (D_LAT - 1)] = tanhf(agg1[i] + xw1[i] * di * di + b4[0]);
    }
}

// ---------------- global_sort_pool: stable top-30 by last channel, per graph ----------
__global__ void sortpool(const float* __restrict__ xcat, float* __restrict__ top) {
    __shared__ float key[NPG];
    __shared__ int   sel[K_TOP];
    int g = blockIdx.x, t = threadIdx.x;
    const float* base = xcat + (long long)g * NPG * D_LAT;
    if (t < NPG) key[t] = base[t * D_LAT + (D_LAT - 1)];
    __syncthreads();
    if (t < NPG) {
        float v = key[t];
        int rank = 0;
        for (int j = 0; j < NPG; ++j) {
            float u = key[j];
            rank += (u > v) || (u == v && j < t);   // stable argsort(-v) tie-break
        }
        if (rank < K_TOP) sel[rank] = t;            // ranks unique -> each slot set once
    }
    __syncthreads();
    float* dst = top + (long long)g * K_TOP * D_LAT;
    for (int idx = t; idx < K_TOP * D_LAT; idx += blockDim.x) {
        int slot = idx / D_LAT, d = idx - slot * D_LAT;
        dst[idx] = base[sel[slot] * D_LAT + d];
    }
}

// ---------------- Conv1d(1,16,97,s=97)+ReLU fused with MaxPool1d(2,2) ----------------
__global__ void conv1_pool(const float* __restrict__ top, const float* __restrict__ Wc1,
                           const float* __restrict__ bc1, float* __restrict__ pool) {
    __shared__ float y1[16 * K_TOP];
    int g = blockIdx.x, t = threadIdx.x;
    const float* tb = top + (long long)g * K_TOP * D_LAT;
    if (t < 16 * K_TOP) {
        int c = t / K_TOP, k = t - c * K_TOP;
        float s = bc1[c];
        for (int d = 0; d < D_LAT; ++d) s += tb[k * D_LAT + d] * Wc1[c * D_LAT + d];
        y1[c * K_TOP + k] = s > 0.0f ? s : 0.0f;
    }
    __syncthreads();
    if (t < 16 * 15) {
        int c = t / 15, j = t - c * 15;
        float a = y1[c * K_TOP + 2 * j], b = y1[c * K_TOP + 2 * j + 1];
        pool[(long long)g * 240 + t] = a > b ? a : b;
    }
}

// ---------------- Conv1d(16,32,5)+ReLU, flattened PyTorch order ----------------
__global__ void conv2k(const float* __restrict__ pool, const float* __restrict__ Wc2,
                       const float* __restrict__ bc2, float* __restrict__ y2) {
    __shared__ float p[240];
    int g = blockIdx.x, t = threadIdx.x;
    if (t < 240) p[t] = pool[(long long)g * 240 + t];
    __syncthreads();
    if (t < 352) {
        int o = t / 11, tt = t - o * 11;
        float s = bc2[o];
#pragma unroll
        for (int i = 0; i < 16; ++i)
#pragma unroll
            for (int k = 0; k < 5; ++k)
                s += p[i * 15 + tt + k] * Wc2[(o * 16 + i) * 5 + k];
        y2[(long long)g * 352 + t] = s > 0.0f ? s : 0.0f;   // t == o*11+tt == flat index
    }
}

// ---------------- FC1 via WMMA: relu([1000,352] @ [352,128] + b) ----------------
// 8 waves per block = the 8 column tiles (128 cols); one 16-row strip per block.
__global__ __launch_bounds__(256)
void fc1_wmma(const float* __restrict__ A, const float* __restrict__ Wl1,
              const float* __restrict__ bl1, float* __restrict__ out, int M) {
    int wave = threadIdx.x >> 5;          // n-tile 0..7
    int lane = threadIdx.x & 31;
    int l15 = lane & 15, half = lane >> 4;
    int rowBase = blockIdx.x * 16;
    int rowA = rowBase + l15; if (rowA >= M) rowA = M - 1;   // clamp loads only
    int ncol = wave * 16 + l15;
    v8f c = {};
    for (int s = 0; s < 88; ++s) {
        int k = 4 * s + 2 * half;
        v2f a = *(const v2f*)(A + rowA * 352 + k);           // 8B-aligned (k even)
        v2f b; b.x = Wl1[k * 128 + ncol];  b.y = Wl1[(k + 1) * 128 + ncol];
        c = __builtin_amdgcn_wmma_f32_16x16x4_f32(false, a, false, b, (short)0, c, false, false);
    }
    float bias = bl1[ncol];
    if (rowBase + 16 <= M) {              // wave-uniform full-tile fast path
#pragma unroll
        for (int r = 0; r < 8; ++r) {
            float v = c[r] + bias;
            out[(rowBase + r + 8 * half) * 128 + ncol] = v > 0.0f ? v : 0.0f;
        }
    } else {                              // only the single partial row-block takes this
#pragma unroll
        for (int r = 0; r < 8; ++r) {
            int row = rowBase + r + 8 * half;
            if (row < M) {
                float v = c[r] + bias;
                out[row * 128 + ncol] = v > 0.0f ? v : 0.0f;
            }
        }
    }
}

// ---------------- FC2: [1000,128] @ [128,1] + b ----------------
__global__ void fc2(const float* __restrict__ x, const float* __restrict__ Wl2,
                    const float* __restrict__ bl2, float* __restrict__ out) {
    int b = blockIdx.x * blockDim.x + threadIdx.x;
    if (b < N_GRAPHS) {
        float s = bl2[0];
#pragma unroll 8
        for (int j = 0; j < 128; ++j) s += x[b * 128 + j] * Wl2[j];
        out[b] = s;
    }
}

// ---------------- host ----------------
extern "C" void kernel_launch(void* const* d_in, const int* in_sizes, int n_in,
                              void* d_out, int out_size, void* d_ws, size_t ws_size,
                              hipStream_t stream) {
    (void)in_sizes; (void)n_in; (void)out_size; (void)ws_size;

    const int*   z    = (const int*)d_in[0];
    const int*   ei   = (const int*)d_in[1];   // [2, E] flat
    // d_in[2] = batch (layout implied: arange(N)//100, unused)
    const float* ew   = (const float*)d_in[3];
    const float* zemb = (const float*)d_in[4];
    const float* Wg[3] = { (const float*)d_in[5], (const float*)d_in[7], (const float*)d_in[9] };
    const float* bg[3] = { (const float*)d_in[6], (const float*)d_in[8], (const float*)d_in[10] };
    const float* Wg4 = (const float*)d_in[11];
    const float* bg4 = (const float*)d_in[12];
    const float* Wc1 = (const float*)d_in[13];
    const float* bc1 = (const float*)d_in[14];
    const float* Wc2 = (const float*)d_in[15];
    const float* bc2 = (const float*)d_in[16];
    const float* Wl1 = (const float*)d_in[17];
    const float* bl1 = (const float*)d_in[18];
    const float* Wl2 = (const float*)d_in[19];
    const float* bl2 = (const float*)d_in[20];
    float* outp = (float*)d_out;

    // workspace carve-out (~105 MB of f32)
    float* w = (float*)d_ws;
    float* dinv = w;  w += N_NODES;
    float* xcat = w;  w += (size_t)N_NODES * D_LAT;
    float* hA   = w;  w += (size_t)N_NODES * HID;
    float* hB   = w;  w += (size_t)N_NODES * HID;
    float* xw   = w;  w += (size_t)N_NODES * HID;   // also reused as xw1 [N]
    float* agg  = w;  w += (size_t)N_NODES * HID;   // also reused as agg1 [N]
    float* top  = w;  w += (size_t)N_GRAPHS * K_TOP * D_LAT;
    float* pool = w;  w += (size_t)N_GRAPHS * 240;
    float* y2   = w;  w += (size_t)N_GRAPHS * 352;
    float* fc1o = w;  w += (size_t)N_GRAPHS * 128;

    const int T = 256;

    // degree / symmetric normalization
    zero_f32<<<cdiv(N_NODES, T), T, 0, stream>>>(dinv, N_NODES);
    deg_accum<<<cdiv(N_EDGES, T), T, 0, stream>>>(ei, ew, dinv);
    deg_to_dinv<<<cdiv(N_NODES, T), T, 0, stream>>>(dinv);

    // x = z_emb[z]
    embed<<<cdiv((long long)N_NODES * HID, T), T, 0, stream>>>(z, zemb, hA);

    // three 32->32 GCN layers
    float* hin = hA; float* hout = hB;
    const int gemmBlocks = cdiv((long long)N_NODES, 128);   // 8 waves x 16 rows per block
    for (int l = 0; l < 3; ++l) {
        gemm_n32_wmma<<<gemmBlocks, 256, 0, stream>>>(hin, Wg[l], xw, N_NODES);
        zero_f32<<<cdiv((long long)N_NODES * HID, T), T, 0, stream>>>(agg, N_NODES * HID);
        scatter32<<<cdiv((long long)N_EDGES * 32, T), T, 0, stream>>>(ei, ew, dinv, xw, agg);
        finalize32<<<cdiv((long long)N_NODES * HID, T), T, 0, stream>>>(agg, xw, dinv, bg[l],
                                                                        hout, xcat, l * 32);
        float* tmp = hin; hin = hout; hout = tmp;
    }

    // 32 -> 1 layer (writes the sort key channel x_cat[:,96])
    gemv_out1<<<cdiv(N_NODES, T), T, 0, stream>>>(hin, Wg4, xw);
    zero_f32<<<cdiv(N_NODES, T), T, 0, stream>>>(agg, N_NODES);
    scatter1<<<cdiv(N_EDGES, T), T, 0, stream>>>(ei, ew, dinv, xw, agg);
    finalize1<<<cdiv(N_NODES, T), T, 0, stream>>>(agg, xw, dinv, bg4, xcat);

    // sort-pool + conv tail
    sortpool<<<N_GRAPHS, 128, 0, stream>>>(xcat, top);
    conv1_pool<<<N_GRAPHS, 512, 0, stream>>>(top, Wc1, bc1, pool);
    conv2k<<<N_GRAPHS, 384, 0, stream>>>(pool, Wc2, bc2, y2);
    fc1_wmma<<<cdiv(N_GRAPHS, 16), 256, 0, stream>>>(y2, Wl1, bl1, fc1o, N_GRAPHS);
    fc2<<<cdiv(N_GRAPHS, T), T, 0, stream>>>(fc1o, Wl2, bl2, outp);
}